// GCN_11836929867924
// MI455X (gfx1250) — compile-verified
//
#include <hip/hip_runtime.h>
#include <math.h>

typedef __attribute__((ext_vector_type(2))) float v2f;
typedef __attribute__((ext_vector_type(8))) float v8f;

#define NODES_PER_TILE 16
#define KDIM 128

// ---------------------------------------------------------------- utilities
__global__ void k_zero(float* __restrict__ p, int n) {
    int i = blockIdx.x * blockDim.x + threadIdx.x;
    if (i < n) p[i] = 0.0f;
}

__global__ void k_deg(const int* __restrict__ src, const int* __restrict__ dst,
                      float* __restrict__ outdeg, float* __restrict__ indeg, int nE) {
    int e = blockIdx.x * blockDim.x + threadIdx.x;
    if (e < nE) {
        unsafeAtomicAdd(&outdeg[src[e]], 1.0f);
        unsafeAtomicAdd(&indeg[dst[e]], 1.0f);
    }
}

// x -> (max(x,1))^-0.5  in place (applied to lnorm and rnorm jointly)
__global__ void k_invsqrt_clip(float* __restrict__ p, int n) {
    int i = blockIdx.x * blockDim.x + threadIdx.x;
    if (i < n) p[i] = rsqrtf(fmaxf(p[i], 1.0f));
}

__global__ void k_relu(float* __restrict__ p, int n) {
    int i = blockIdx.x * blockDim.x + threadIdx.x;
    if (i < n) p[i] = fmaxf(p[i], 0.0f);
}

// x[row,f] = relu(x[row,f] * rnorm[row]) for F=128 (row = i>>7)
__global__ void k_relu_scale(float* __restrict__ p, const float* __restrict__ rnorm, int n) {
    int i = blockIdx.x * blockDim.x + threadIdx.x;
    if (i < n) p[i] = fmaxf(p[i] * rnorm[i >> 7], 0.0f);
}

// ---------------------------------------------------------------- WMMA GEMM
// C[N_rows, NC] = (rowscale ? diag(scale) : I) * A[N_rows,128] @ W[128,NC]
// One block = 128 threads (4 waves) = one 16-row strip. Each wave owns
// column tiles t = wave, wave+4, ...  K=128 -> 32 chained V_WMMA_F32_16X16X4_F32.
template <int NC>
__global__ __launch_bounds__(128) void k_gemm_wmma(const float* __restrict__ A,
                                                   const float* __restrict__ W,
                                                   const float* __restrict__ scale,
                                                   float* __restrict__ C) {
    __shared__ float lA[NODES_PER_TILE * KDIM];  // 8 KB

    const int row0 = blockIdx.x * NODES_PER_TILE;

    // Stage the 16x128 A-strip in LDS, fusing the optional per-row scale.
    for (int i = threadIdx.x; i < (NODES_PER_TILE * KDIM) / 4; i += 128) {
        int r  = i >> 5;   // row within strip (32 float4 per row)
        int c4 = i & 31;   // float4 column
        float4 v = reinterpret_cast<const float4*>(A + (size_t)(row0 + r) * KDIM)[c4];
        float s = scale ? scale[row0 + r] : 1.0f;
        v.x *= s; v.y *= s; v.z *= s; v.w *= s;
        reinterpret_cast<float4*>(lA)[i] = v;
    }
    __syncthreads();

    const int wave = threadIdx.x >> 5;
    const int lane = threadIdx.x & 31;
    const int l16  = lane & 15;
    const int hi   = lane >> 4;           // 0: K pair {0,1}, 1: K pair {2,3}
    const int NT   = (NC + 15) / 16;

    // A fragment source: row l16, consecutive K pair selected by hi
    const float* arow = lA + l16 * KDIM + 2 * hi;

    for (int t = wave; t < NT; t += 4) {
        const int n0  = t * 16;
        const int col = n0 + l16;
        const bool colOK = (NC % 16 == 0) || (col < NC);

        v8f acc = {};
#pragma unroll
        for (int kk = 0; kk < KDIM / 4; ++kk) {
            const int kb = kk * 4;
            v2f a;
            a.x = arow[kb];
            a.y = arow[kb + 1];
            const int kr = kb + 2 * hi;
            v2f b;
            if (colOK) {
                b.x = W[(size_t)kr * NC + col];
                b.y = W[(size_t)(kr + 1) * NC + col];
            } else {
                b.x = 0.0f; b.y = 0.0f;
            }
            // D = A(16x4,f32) x B(4x16,f32) + C ; full-precision fp32 matrix pipe
            acc = __builtin_amdgcn_wmma_f32_16x16x4_f32(
                false, a, false, b, (short)0, acc, false, false);
        }

        if (colOK) {
#pragma unroll
            for (int r = 0; r < 8; ++r) {
                const int m = r + 8 * hi;  // C/D layout: VGPR r -> row r / r+8
                C[(size_t)(row0 + m) * NC + col] = acc[r];
            }
        }
    }
}

// ---------------------------------------------------------------- scatter agg
// F=128: one wave per edge. Edge index is wave-uniform -> scalarize it so the
// src/dst fetches become s_load and the row base addresses live in SGPRs
// (saddr-form global ops); lanes cover the 128 floats as 32 float4 groups.
__global__ void k_agg128(const float* __restrict__ X, const int* __restrict__ src,
                         const int* __restrict__ dst, float* __restrict__ out, int nE) {
    int e = blockIdx.x * (blockDim.x >> 5) + (threadIdx.x >> 5);
    if (e >= nE) return;                       // whole-wave exit
    e = __builtin_amdgcn_readfirstlane(e);     // provably uniform
    const int lane = threadIdx.x & 31;
    const int s = src[e];                      // scalar loads
    const int d = dst[e];
    const float4 v = reinterpret_cast<const float4*>(X + (size_t)s * 128)[lane];
    float* o = out + (size_t)d * 128 + lane * 4;
    unsafeAtomicAdd(o + 0, v.x);
    unsafeAtomicAdd(o + 1, v.y);
    unsafeAtomicAdd(o + 2, v.z);
    unsafeAtomicAdd(o + 3, v.w);
}

// F=40: one thread per (edge, float4 group), G=10 groups per edge.
__global__ void k_agg40(const float* __restrict__ X, const int* __restrict__ src,
                        const int* __restrict__ dst, float* __restrict__ out, int nE) {
    const int G = 10;
    long long tid = (long long)blockIdx.x * blockDim.x + threadIdx.x;
    if (tid >= (long long)nE * G) return;
    int e = (int)(tid / G);
    int g = (int)(tid % G);
    int s = src[e], d = dst[e];
    float4 v = reinterpret_cast<const float4*>(X + (size_t)s * 40)[g];
    float* o = out + (size_t)d * 40 + g * 4;
    unsafeAtomicAdd(o + 0, v.x);
    unsafeAtomicAdd(o + 1, v.y);
    unsafeAtomicAdd(o + 2, v.z);
    unsafeAtomicAdd(o + 3, v.w);
}

// ---------------------------------------------------------------- finalize
// out[row,:] = log_softmax(out[row,:] * rnorm[row] + b2) ; one wave per row (40 cols)
__global__ void k_finalize(float* __restrict__ out, const float* __restrict__ rnorm,
                           const float* __restrict__ b2, int nrows) {
    int row = blockIdx.x * (blockDim.x >> 5) + (threadIdx.x >> 5);
    if (row >= nrows) return;
    int lane = threadIdx.x & 31;
    float rn = rnorm[row];
    float* p = out + (size_t)row * 40;

    float a = p[lane] * rn + b2[lane];
    float b = (lane < 8) ? (p[lane + 32] * rn + b2[lane + 32]) : -__builtin_inff();

    float m = fmaxf(a, b);
#pragma unroll
    for (int off = 16; off > 0; off >>= 1) m = fmaxf(m, __shfl_xor(m, off, 32));

    float se = expf(a - m) + ((lane < 8) ? expf(b - m) : 0.0f);
#pragma unroll
    for (int off = 16; off > 0; off >>= 1) se += __shfl_xor(se, off, 32);

    float lse = m + logf(se);
    p[lane] = a - lse;
    if (lane < 8) p[lane + 32] = b - lse;
}

// ---------------------------------------------------------------- launcher
extern "C" void kernel_launch(void* const* d_in, const int* in_sizes, int n_in,
                              void* d_out, int out_size, void* d_ws, size_t ws_size,
                              hipStream_t stream) {
    const float* feat = (const float*)d_in[0];
    const int*   src  = (const int*)d_in[1];
    const int*   dst  = (const int*)d_in[2];
    const float* W1   = (const float*)d_in[3];
    const float* Wh   = (const float*)d_in[4];
    const float* W2   = (const float*)d_in[5];
    const float* b2   = (const float*)d_in[6];
    float* out = (float*)d_out;

    const int N = in_sizes[0] / 128;   // 100000 (divisible by 16)
    const int E = in_sizes[1];         // 1600000

    float* lnorm = (float*)d_ws;                    // N
    float* rnorm = lnorm + N;                       // N
    float* bufA  = rnorm + N;                       // N*128
    float* bufB  = bufA + (size_t)N * 128;          // N*128

    const dim3 B256(256);
    auto nb = [](long long n) { return dim3((unsigned)((n + 255) / 256)); };

    // degrees -> lnorm (out-deg^-1/2), rnorm (in-deg^-1/2)
    k_zero<<<nb(2LL * N), B256, 0, stream>>>(lnorm, 2 * N);
    k_deg<<<nb(E), B256, 0, stream>>>(src, dst, lnorm, rnorm, E);
    k_invsqrt_clip<<<nb(2LL * N), B256, 0, stream>>>(lnorm, 2 * N);

    // layer 1: bufA = X @ W1 ; bufB = relu(agg(bufA))
    k_gemm_wmma<128><<<dim3(N / 16), dim3(128), 0, stream>>>(feat, W1, nullptr, bufA);
    k_zero<<<nb((long long)N * 128), B256, 0, stream>>>(bufB, N * 128);
    k_agg128<<<dim3((E + 7) / 8), B256, 0, stream>>>(bufA, src, dst, bufB, E);
    k_relu<<<nb((long long)N * 128), B256, 0, stream>>>(bufB, N * 128);

    // layer 2: bufA = (H1 * lnorm) @ Wh ; bufB = relu(agg(bufA) * rnorm)
    k_gemm_wmma<128><<<dim3(N / 16), dim3(128), 0, stream>>>(bufB, Wh, lnorm, bufA);
    k_zero<<<nb((long long)N * 128), B256, 0, stream>>>(bufB, N * 128);
    k_agg128<<<dim3((E + 7) / 8), B256, 0, stream>>>(bufA, src, dst, bufB, E);
    k_relu_scale<<<nb((long long)N * 128), B256, 0, stream>>>(bufB, rnorm, N * 128);

    // layer 3: bufA[:, :40] = (H2 * lnorm) @ W2 ; out = agg(...) * rnorm + b2 ; log_softmax
    k_gemm_wmma<40><<<dim3(N / 16), dim3(128), 0, stream>>>(bufB, W2, lnorm, bufA);
    k_zero<<<nb((long long)N * 40), B256, 0, stream>>>(out, N * 40);
    k_agg40<<<nb((long long)E * 10), B256, 0, stream>>>(bufA, src, dst, out, E);
    k_finalize<<<dim3((N + 7) / 8), B256, 0, stream>>>(out, rnorm, b2, N);
}